// SelectiveSSM_34102040330334
// MI455X (gfx1250) — compile-verified
//
#include <hip/hip_runtime.h>
#include <hip/hip_bf16.h>

// Problem constants
#define B_SZ   2
#define SEQ    2048
#define DM     1024     // d_model
#define DS     16       // d_state
#define DTR    64       // dt_rank
#define ML     (B_SZ*SEQ)   // 4096 rows for the GEMMs
#define NC     32       // number of scan chunks
#define CL     (SEQ/NC) // 64 timesteps per chunk
#define NCHAIN (B_SZ*DM) // 2048 scan chains
#define LOG2E  1.44269504088896340736f
#define PFD    8        // prefetch distance (timesteps)

typedef float v2f __attribute__((ext_vector_type(2)));
typedef float v8f __attribute__((ext_vector_type(8)));

// ---------------------------------------------------------------------------
// Kernel 1: dt = softplus(dtin @ dt_W^T + 2*dt_b)
//   M=4096 (b*l), N=1024 (d), K=64 (r).  fp32 WMMA 16x16x4, one wave per tile.
// ---------------------------------------------------------------------------
__global__ __launch_bounds__(32)
void k_dt_gemm(const float* __restrict__ dtin,  // (ML, DTR)
               const float* __restrict__ W,     // (DM, DTR)
               const float* __restrict__ bias,  // (DM)
               float* __restrict__ dtout)       // (ML, DM)
{
    const int lane = threadIdx.x & 31;
    const int tm = blockIdx.x;          // M tile: 0..255
    const int tn = blockIdx.y;          // N tile: 0..63
    const int half = lane >> 4;         // 0: K=0,1  1: K=2,3
    const int l16  = lane & 15;
    const int rowA = tm * 16 + l16;     // A-matrix row
    const int colB = tn * 16 + l16;     // B-matrix col

    v8f acc = {};
    #pragma unroll 4
    for (int kk = 0; kk < DTR; kk += 4) {
        const int ka = kk + half * 2;
        v2f a = *(const v2f*)(dtin + rowA * DTR + ka);  // A[m][k], A[m][k+1]
        v2f b = *(const v2f*)(W    + colB * DTR + ka);  // B[k][n] = W[n][k]
        acc = __builtin_amdgcn_wmma_f32_16x16x4_f32(
                  false, a, false, b, (short)0, acc, false, false);
    }

    const int colD = tn * 16 + l16;
    const float bb = 2.0f * bias[colD];
    #pragma unroll
    for (int r = 0; r < 8; ++r) {
        const int rowD = tm * 16 + r + half * 8;   // D layout: M = r + 8*(lane>=16)
        const float z = acc[r] + bb;
        // stable & precise softplus (dt is often ~e^-17; keep log1pf accuracy)
        const float sp = fmaxf(z, 0.0f) + log1pf(__expf(-fabsf(z)));
        dtout[rowD * DM + colD] = sp;
    }
}

// ---------------------------------------------------------------------------
// Kernel 2: Bm = x @ B_W^T, Cm = x @ C_W^T
//   M=4096, N=16 (single tile), K=1024.  Shared A loads, two WMMAs per k-step.
// ---------------------------------------------------------------------------
__global__ __launch_bounds__(32)
void k_bc_gemm(const float* __restrict__ x,   // (ML, DM)
               const float* __restrict__ BW,  // (DS, DM)
               const float* __restrict__ CW,  // (DS, DM)
               float* __restrict__ Bm,        // (ML, DS)
               float* __restrict__ Cm)        // (ML, DS)
{
    const int lane = threadIdx.x & 31;
    const int tm = blockIdx.x;          // 0..255
    const int half = lane >> 4;
    const int l16  = lane & 15;
    const int rowA = tm * 16 + l16;

    v8f accB = {};
    v8f accC = {};
    #pragma unroll 4
    for (int kk = 0; kk < DM; kk += 4) {
        const int ka = kk + half * 2;
        v2f a  = *(const v2f*)(x  + rowA * DM + ka);
        v2f bb = *(const v2f*)(BW + l16  * DM + ka);   // B[k][n] = B_W[n][k]
        v2f cc = *(const v2f*)(CW + l16  * DM + ka);
        accB = __builtin_amdgcn_wmma_f32_16x16x4_f32(
                   false, a, false, bb, (short)0, accB, false, false);
        accC = __builtin_amdgcn_wmma_f32_16x16x4_f32(
                   false, a, false, cc, (short)0, accC, false, false);
    }

    #pragma unroll
    for (int r = 0; r < 8; ++r) {
        const int rowD = tm * 16 + r + half * 8;
        Bm[rowD * DS + l16] = accB[r];
        Cm[rowD * DS + l16] = accC[r];
    }
}

// ---------------------------------------------------------------------------
// Kernel 3: scan pass 1 — per (chain, chunk): decay product P[n] and local
// end-state Hend[n] starting from h=0.  grid (DM/256, B_SZ, NC), block 256.
// ---------------------------------------------------------------------------
__global__ __launch_bounds__(256)
void k_scan_pass1(const float* __restrict__ dt,   // (ML, DM)
                  const float* __restrict__ x,    // (ML, DM)
                  const float* __restrict__ Bm,   // (ML, DS)
                  const float* __restrict__ A_log,// (DS)
                  float* __restrict__ Pbuf,       // (NC, NCHAIN, DS)
                  float* __restrict__ Hbuf)       // (NC, NCHAIN, DS)
{
    const int d = blockIdx.x * 256 + threadIdx.x;
    const int b = blockIdx.y;
    const int c = blockIdx.z;
    const int t0 = c * CL;

    __shared__ float Bs[CL][DS];
    for (int i = threadIdx.x; i < CL * DS; i += 256)
        Bs[i >> 4][i & 15] = Bm[(b * SEQ + t0) * DS + i];
    __syncthreads();

    // A2[n] = A[n] * log2(e), so dA = exp2(dt * A2[n]) — one v_mul + one v_exp
    float A2[DS];
    #pragma unroll
    for (int n = 0; n < DS; ++n) A2[n] = -__expf(A_log[n]) * LOG2E;

    float h[DS], P[DS];
    #pragma unroll
    for (int n = 0; n < DS; ++n) { h[n] = 0.0f; P[n] = 1.0f; }

    const float* dtp = dt + (size_t)(b * SEQ + t0) * DM + d;
    const float* xp  = x  + (size_t)(b * SEQ + t0) * DM + d;

    for (int t = 0; t < CL; ++t) {
        __builtin_prefetch(dtp + (t + PFD) * DM, 0, 1);   // global_prefetch_b8
        __builtin_prefetch(xp  + (t + PFD) * DM, 0, 1);
        const float dtv = dtp[t * DM];
        const float dtx = dtv * xp[t * DM];
        #pragma unroll
        for (int n = 0; n < DS; ++n) {
            const float e = __builtin_amdgcn_exp2f(dtv * A2[n]);
            h[n] = e * h[n] + dtx * Bs[t][n];
            P[n] *= e;
        }
    }

    const int chain = b * DM + d;
    #pragma unroll
    for (int n = 0; n < DS; ++n) {
        Pbuf[((size_t)c * NCHAIN + chain) * DS + n] = P[n];
        Hbuf[((size_t)c * NCHAIN + chain) * DS + n] = h[n];
    }
}

// ---------------------------------------------------------------------------
// Kernel 4: combine — sequential prefix over the NC=32 chunks per chain.
// Converts Hbuf in place from "chunk-local end state" to "incoming state".
// ---------------------------------------------------------------------------
__global__ __launch_bounds__(256)
void k_combine(const float* __restrict__ Pbuf,
               float* __restrict__ Hbuf)
{
    const int chain = blockIdx.x * 256 + threadIdx.x;   // 0..2047
    float S[DS];
    #pragma unroll
    for (int n = 0; n < DS; ++n) S[n] = 0.0f;

    for (int c = 0; c < NC; ++c) {
        #pragma unroll
        for (int n = 0; n < DS; ++n) {
            const size_t idx = ((size_t)c * NCHAIN + chain) * DS + n;
            const float p    = Pbuf[idx];
            const float hend = Hbuf[idx];
            Hbuf[idx] = S[n];              // incoming state for chunk c
            S[n] = p * S[n] + hend;        // state entering chunk c+1
        }
    }
}

// ---------------------------------------------------------------------------
// Kernel 5: scan pass 2 — re-run each chunk from its correct incoming state,
// produce y[t] = sum_n h[t,n]*Cm[t,n] + x*D_vec.
// ---------------------------------------------------------------------------
__global__ __launch_bounds__(256)
void k_scan_pass2(const float* __restrict__ dt,
                  const float* __restrict__ x,
                  const float* __restrict__ Bm,
                  const float* __restrict__ Cm,
                  const float* __restrict__ A_log,
                  const float* __restrict__ Dvec,
                  const float* __restrict__ Hbuf,  // incoming states
                  float* __restrict__ out)         // (ML, DM)
{
    const int d = blockIdx.x * 256 + threadIdx.x;
    const int b = blockIdx.y;
    const int c = blockIdx.z;
    const int t0 = c * CL;

    __shared__ float Bs[CL][DS];
    __shared__ float Cs[CL][DS];
    for (int i = threadIdx.x; i < CL * DS; i += 256) {
        Bs[i >> 4][i & 15] = Bm[(b * SEQ + t0) * DS + i];
        Cs[i >> 4][i & 15] = Cm[(b * SEQ + t0) * DS + i];
    }
    __syncthreads();

    float A2[DS];
    #pragma unroll
    for (int n = 0; n < DS; ++n) A2[n] = -__expf(A_log[n]) * LOG2E;

    const int chain = b * DM + d;
    float h[DS];
    #pragma unroll
    for (int n = 0; n < DS; ++n)
        h[n] = Hbuf[((size_t)c * NCHAIN + chain) * DS + n];

    const float dv = Dvec[d];
    const float* dtp = dt  + (size_t)(b * SEQ + t0) * DM + d;
    const float* xp  = x   + (size_t)(b * SEQ + t0) * DM + d;
    float* op        = out + (size_t)(b * SEQ + t0) * DM + d;

    for (int t = 0; t < CL; ++t) {
        __builtin_prefetch(dtp + (t + PFD) * DM, 0, 1);
        __builtin_prefetch(xp  + (t + PFD) * DM, 0, 1);
        const float dtv = dtp[t * DM];
        const float xv  = xp[t * DM];
        const float dtx = dtv * xv;
        float y = 0.0f;
        #pragma unroll
        for (int n = 0; n < DS; ++n) {
            const float e = __builtin_amdgcn_exp2f(dtv * A2[n]);
            h[n] = e * h[n] + dtx * Bs[t][n];
            y += h[n] * Cs[t][n];
        }
        op[t * DM] = y + xv * dv;
    }
}

// ---------------------------------------------------------------------------
// Launcher
// ---------------------------------------------------------------------------
extern "C" void kernel_launch(void* const* d_in, const int* in_sizes, int n_in,
                              void* d_out, int out_size, void* d_ws, size_t ws_size,
                              hipStream_t stream) {
    (void)in_sizes; (void)n_in; (void)out_size; (void)ws_size;

    const float* x     = (const float*)d_in[0];  // (B,SEQ,DM)
    const float* dtin  = (const float*)d_in[1];  // (B,SEQ,DTR)
    const float* A_log = (const float*)d_in[2];  // (DS)
    const float* dt_W  = (const float*)d_in[3];  // (DM,DTR)
    const float* dt_b  = (const float*)d_in[4];  // (DM)
    const float* B_W   = (const float*)d_in[5];  // (DS,DM)
    const float* C_W   = (const float*)d_in[6];  // (DS,DM)
    const float* Dvec  = (const float*)d_in[7];  // (DM)
    float* out = (float*)d_out;

    // Workspace layout (bytes)
    char* ws = (char*)d_ws;
    float* dtbuf = (float*)(ws);                                   // ML*DM floats
    float* Bm    = (float*)(ws + (size_t)ML * DM * 4);             // ML*DS
    float* Cm    = (float*)(ws + (size_t)ML * DM * 4 + (size_t)ML * DS * 4);
    float* Pbuf  = (float*)(ws + (size_t)ML * DM * 4 + (size_t)2 * ML * DS * 4);
    float* Hbuf  = (float*)(ws + (size_t)ML * DM * 4 + (size_t)2 * ML * DS * 4
                               + (size_t)NC * NCHAIN * DS * 4);

    // 1) dt projection + softplus (WMMA f32)
    k_dt_gemm<<<dim3(ML / 16, DM / 16), 32, 0, stream>>>(dtin, dt_W, dt_b, dtbuf);

    // 2) B and C projections (WMMA f32)
    k_bc_gemm<<<dim3(ML / 16), 32, 0, stream>>>(x, B_W, C_W, Bm, Cm);

    // 3) chunked scan, pass 1: per-chunk decay products + local end states
    k_scan_pass1<<<dim3(DM / 256, B_SZ, NC), 256, 0, stream>>>(
        dtbuf, x, Bm, A_log, Pbuf, Hbuf);

    // 4) combine chunk states (32-step sequential prefix per chain)
    k_combine<<<dim3(NCHAIN / 256), 256, 0, stream>>>(Pbuf, Hbuf);

    // 5) pass 2: re-scan each chunk from correct incoming state, emit y
    k_scan_pass2<<<dim3(DM / 256, B_SZ, NC), 256, 0, stream>>>(
        dtbuf, x, Bm, Cm, A_log, Dvec, Hbuf, out);
}